// ColorNet_emb_19713899889091
// MI455X (gfx1250) — compile-verified
//
#include <hip/hip_runtime.h>
#include <hip/hip_bf16.h>

// ---------------------------------------------------------------------------
// Bipartite GNN (Gasse-style) for MI455X / gfx1250, wave32.
//
// Key rewrites vs the reference (exact math, not approximations):
//  * LN over a size-1 axis == e_lnb, so the edge-feature term folds into the
//    bias of the target-side linear:  bias = bl[k] + e_lnb * We[k].
//  * segsum(relu(LN(pre)) @ Wf + bf) == segsum(relu(LN(pre))) @ Wf + deg*bf,
//    moving the 64x64 GEMM from 500k edges to 25k/50k nodes.
//  * All shared-weight N x 64 @ 64 x 64 GEMMs run on V_WMMA_F32_16X16X4_F32.
// ---------------------------------------------------------------------------

#define NV 50000
#define NC 25000
#define NE 500000
#define VP 50048   // NV padded to 128 (one 256-thread block = 8 waves = 128 rows)
#define CP 25088   // NC padded to 128
#define EPS_C 1e-5f

typedef __attribute__((ext_vector_type(2))) float v2f;
typedef __attribute__((ext_vector_type(8))) float v8f;

__device__ __forceinline__ v8f wmma4(v2f a, v2f b, v8f c) {
  // D = A(16x4 f32) x B(4x16 f32) + C(16x16 f32)
  return __builtin_amdgcn_wmma_f32_16x16x4_f32(false, a, false, b, (short)0, c,
                                               false, false);
}

// ---------------------------------------------------------------------------
// degree count (float, feeds deg*bf)
// ---------------------------------------------------------------------------
__global__ __launch_bounds__(256) void k_deg(const int* __restrict__ idx,
                                             float* __restrict__ deg, int ne) {
  int e = blockIdx.x * 256 + threadIdx.x;
  if (e < ne) atomicAdd(&deg[idx[e]], 1.0f);
}

// ---------------------------------------------------------------------------
// variable expert embed + group embed  (wave32 shuffle-broadcast MLP)
// one wave per node; lane handles output channels {lane, lane+32}
// ---------------------------------------------------------------------------
__global__ __launch_bounds__(256) void k_embed_var(
    const float* __restrict__ xf, const float* __restrict__ gf,
    const int* __restrict__ color, const float* __restrict__ lng,
    const float* __restrict__ lnb, const float* __restrict__ W1,
    const float* __restrict__ b1, const float* __restrict__ W2,
    const float* __restrict__ b2, const float* __restrict__ g_lng,
    const float* __restrict__ g_lnb, const float* __restrict__ g_W,
    const float* __restrict__ g_b, float* __restrict__ out, int n) {
  int lane = threadIdx.x & 31;
  int node = blockIdx.x * 8 + (threadIdx.x >> 5);
  if (node >= n) return;
  int c = color[node];
  float xv[7], m = 0.f;
#pragma unroll
  for (int f = 0; f < 7; ++f) { xv[f] = xf[(size_t)node * 7 + f]; m += xv[f]; }
  m *= (1.f / 7.f);
  float vv = 0.f;
#pragma unroll
  for (int f = 0; f < 7; ++f) { float d = xv[f] - m; vv += d * d; }
  float inv = rsqrtf(vv * (1.f / 7.f) + EPS_C);
  int o0 = lane, o1 = lane + 32;
  float a0 = b1[c * 64 + o0], a1 = b1[c * 64 + o1];
#pragma unroll
  for (int f = 0; f < 7; ++f) {
    float h = (xv[f] - m) * inv * lng[c * 7 + f] + lnb[c * 7 + f];
    a0 += h * W1[c * 448 + f * 64 + o0];
    a1 += h * W1[c * 448 + f * 64 + o1];
  }
  a0 = fmaxf(a0, 0.f); a1 = fmaxf(a1, 0.f);
  float h0 = b2[c * 64 + o0], h1 = b2[c * 64 + o1];
  const float* W2c = W2 + (size_t)c * 4096;
#pragma unroll
  for (int d = 0; d < 32; ++d) {
    float hd = __shfl(a0, d);
    h0 += hd * W2c[d * 64 + o0];
    h1 += hd * W2c[d * 64 + o1];
  }
#pragma unroll
  for (int d = 0; d < 32; ++d) {
    float hd = __shfl(a1, d);
    h0 += hd * W2c[(d + 32) * 64 + o0];
    h1 += hd * W2c[(d + 32) * 64 + o1];
  }
  h0 = fmaxf(h0, 0.f); h1 = fmaxf(h1, 0.f);
  // group: 32 features == 32 lanes
  float gv = gf[(size_t)node * 32 + lane];
  float gs = gv, gq = gv * gv;
#pragma unroll
  for (int msk = 1; msk < 32; msk <<= 1) {
    gs += __shfl_xor(gs, msk);
    gq += __shfl_xor(gq, msk);
  }
  float gm = gs * (1.f / 32.f);
  float gn = (gv - gm) * rsqrtf(gq * (1.f / 32.f) - gm * gm + EPS_C) *
                 g_lng[lane] + g_lnb[lane];
  float e0 = g_b[o0], e1 = g_b[o1];
#pragma unroll
  for (int l = 0; l < 32; ++l) {
    float gl = __shfl(gn, l);
    e0 += gl * g_W[l * 64 + o0];
    e1 += gl * g_W[l * 64 + o1];
  }
  out[(size_t)node * 64 + o0] = h0 + fmaxf(e0, 0.f);
  out[(size_t)node * 64 + o1] = h1 + fmaxf(e1, 0.f);
}

// constraint expert embed (2 input features)
__global__ __launch_bounds__(256) void k_embed_cons(
    const float* __restrict__ xf, const int* __restrict__ color,
    const float* __restrict__ lng, const float* __restrict__ lnb,
    const float* __restrict__ W1, const float* __restrict__ b1,
    const float* __restrict__ W2, const float* __restrict__ b2,
    float* __restrict__ out, int n) {
  int lane = threadIdx.x & 31;
  int node = blockIdx.x * 8 + (threadIdx.x >> 5);
  if (node >= n) return;
  int c = color[node];
  float x0 = xf[(size_t)node * 2 + 0], x1 = xf[(size_t)node * 2 + 1];
  float m = 0.5f * (x0 + x1);
  float d0 = x0 - m, d1 = x1 - m;
  float inv = rsqrtf(0.5f * (d0 * d0 + d1 * d1) + EPS_C);
  float hn0 = d0 * inv * lng[c * 2 + 0] + lnb[c * 2 + 0];
  float hn1 = d1 * inv * lng[c * 2 + 1] + lnb[c * 2 + 1];
  int o0 = lane, o1 = lane + 32;
  float a0 = fmaxf(b1[c * 64 + o0] + hn0 * W1[c * 128 + o0] +
                       hn1 * W1[c * 128 + 64 + o0], 0.f);
  float a1 = fmaxf(b1[c * 64 + o1] + hn0 * W1[c * 128 + o1] +
                       hn1 * W1[c * 128 + 64 + o1], 0.f);
  float h0 = b2[c * 64 + o0], h1 = b2[c * 64 + o1];
  const float* W2c = W2 + (size_t)c * 4096;
#pragma unroll
  for (int d = 0; d < 32; ++d) {
    float hd = __shfl(a0, d);
    h0 += hd * W2c[d * 64 + o0];
    h1 += hd * W2c[d * 64 + o1];
  }
#pragma unroll
  for (int d = 0; d < 32; ++d) {
    float hd = __shfl(a1, d);
    h0 += hd * W2c[(d + 32) * 64 + o0];
    h1 += hd * W2c[(d + 32) * 64 + o1];
  }
  out[(size_t)node * 64 + o0] = fmaxf(h0, 0.f);
  out[(size_t)node * 64 + o1] = fmaxf(h1, 0.f);
}

// ---------------------------------------------------------------------------
// Y[Npad x 64] = X @ W(64x64) + bias  (bias = b1 + b2scale[0]*b2, any may be 0)
// wave per 16-row tile; WMMA f32 16x16x4; grid = Npad/128, block 256.
// ---------------------------------------------------------------------------
__global__ __launch_bounds__(256) void k_gemm64(
    const float* __restrict__ X, const float* __restrict__ W,
    const float* __restrict__ b1, const float* __restrict__ b2,
    const float* __restrict__ b2scale, float* __restrict__ Y) {
  int lane = threadIdx.x & 31;
  int row0 = (blockIdx.x * 8 + (threadIdx.x >> 5)) * 16;
  int half = lane >> 4, l15 = lane & 15;
  const float2* xrow = (const float2*)(X + (size_t)(row0 + l15) * 64);
  v2f afr[16];
#pragma unroll
  for (int ks = 0; ks < 16; ++ks) {
    float2 t = xrow[2 * ks + half];
    afr[ks].x = t.x; afr[ks].y = t.y;
  }
#pragma unroll
  for (int nt = 0; nt < 4; ++nt) {
    int col = nt * 16 + l15;
    float b = 0.f;
    if (b1) b = b1[col];
    if (b2) b += b2scale[0] * b2[col];
    v8f acc;
#pragma unroll
    for (int v = 0; v < 8; ++v) acc[v] = b;
#pragma unroll
    for (int ks = 0; ks < 16; ++ks) {
      int kk = 4 * ks + 2 * half;
      v2f bf;
      bf.x = W[kk * 64 + col];
      bf.y = W[(kk + 1) * 64 + col];
      acc = wmma4(afr[ks], bf, acc);
    }
    float* yr = Y + (size_t)row0 * 64 + col;
#pragma unroll
    for (int v = 0; v < 8; ++v) yr[(size_t)(v + half * 8) * 64] = acc[v];
  }
}

// ---------------------------------------------------------------------------
// per-edge: t = relu(LN(R[i] + L[j])); S[i] += t   (32 lanes/edge, 2 ch/lane)
// ---------------------------------------------------------------------------
__global__ __launch_bounds__(256) void k_edge(
    const float* __restrict__ R, const float* __restrict__ L,
    const int* __restrict__ tgt, const int* __restrict__ src,
    const float* __restrict__ flng, const float* __restrict__ flnb,
    float* __restrict__ S, int ne) {
  int e = blockIdx.x * 8 + (threadIdx.x >> 5);
  if (e >= ne) return;
  int lane = threadIdx.x & 31;
  int i = tgt[e], j = src[e];
  float2 r = ((const float2*)(R + (size_t)i * 64))[lane];
  float2 l = ((const float2*)(L + (size_t)j * 64))[lane];
  float x0 = r.x + l.x, x1 = r.y + l.y;
  float s = x0 + x1, q = x0 * x0 + x1 * x1;
#pragma unroll
  for (int msk = 1; msk < 32; msk <<= 1) {
    s += __shfl_xor(s, msk);
    q += __shfl_xor(q, msk);
  }
  float mean = s * (1.f / 64.f);
  float inv = rsqrtf(q * (1.f / 64.f) - mean * mean + EPS_C);
  int c0 = lane * 2;
  float t0 = fmaxf((x0 - mean) * inv * flng[c0] + flnb[c0], 0.f);
  float t1 = fmaxf((x1 - mean) * inv * flng[c0 + 1] + flnb[c0 + 1], 0.f);
  atomicAdd(&S[(size_t)i * 64 + c0], t0);
  atomicAdd(&S[(size_t)i * 64 + c0 + 1], t1);
}

// ---------------------------------------------------------------------------
// X1 = LN(S @ Wf + deg*bf) * plng + plnb        (WMMA + in-register LN)
// ---------------------------------------------------------------------------
__global__ __launch_bounds__(256) void k_postagg(
    const float* __restrict__ S, const float* __restrict__ deg,
    const float* __restrict__ Wf, const float* __restrict__ bfv,
    const float* __restrict__ png, const float* __restrict__ pnb,
    float* __restrict__ X1) {
  int lane = threadIdx.x & 31;
  int row0 = (blockIdx.x * 8 + (threadIdx.x >> 5)) * 16;
  int half = lane >> 4, l15 = lane & 15;
  const float2* srow = (const float2*)(S + (size_t)(row0 + l15) * 64);
  v2f afr[16];
#pragma unroll
  for (int ks = 0; ks < 16; ++ks) {
    float2 t = srow[2 * ks + half];
    afr[ks].x = t.x; afr[ks].y = t.y;
  }
  float dg[8];
#pragma unroll
  for (int v = 0; v < 8; ++v) dg[v] = deg[row0 + v + half * 8];
  v8f acc[4];
#pragma unroll
  for (int nt = 0; nt < 4; ++nt) {
    int col = nt * 16 + l15;
    float bc = bfv[col];
#pragma unroll
    for (int v = 0; v < 8; ++v) acc[nt][v] = dg[v] * bc;
#pragma unroll
    for (int ks = 0; ks < 16; ++ks) {
      int kk = 4 * ks + 2 * half;
      v2f bf;
      bf.x = Wf[kk * 64 + col];
      bf.y = Wf[(kk + 1) * 64 + col];
      acc[nt] = wmma4(afr[ks], bf, acc[nt]);
    }
  }
  float gc[4], bc2[4];
#pragma unroll
  for (int nt = 0; nt < 4; ++nt) {
    gc[nt] = png[nt * 16 + l15];
    bc2[nt] = pnb[nt * 16 + l15];
  }
#pragma unroll
  for (int v = 0; v < 8; ++v) {
    float e0 = acc[0][v], e1 = acc[1][v], e2 = acc[2][v], e3 = acc[3][v];
    float s = e0 + e1 + e2 + e3;
    float q = e0 * e0 + e1 * e1 + e2 * e2 + e3 * e3;
#pragma unroll
    for (int msk = 1; msk < 16; msk <<= 1) {  // within-half 16-lane reduce
      s += __shfl_xor(s, msk);
      q += __shfl_xor(q, msk);
    }
    float mean = s * (1.f / 64.f);
    float inv = rsqrtf(q * (1.f / 64.f) - mean * mean + EPS_C);
    float* xr = X1 + (size_t)(row0 + v + half * 8) * 64 + l15;
#pragma unroll
    for (int nt = 0; nt < 4; ++nt)
      xr[nt * 16] = (acc[nt][v] - mean) * inv * gc[nt] + bc2[nt];
  }
}

// ---------------------------------------------------------------------------
// right = relu([X1 ; right] @ Wo1 + bo1) @ Wo2 + bo2   (two WMMA GEMMs,
// LDS transpose between them; in-place on `right`)
// ---------------------------------------------------------------------------
__global__ __launch_bounds__(256) void k_outmlp(
    const float* __restrict__ X1, float* __restrict__ right,
    const float* __restrict__ Wo1, const float* __restrict__ bo1,
    const float* __restrict__ Wo2, const float* __restrict__ bo2) {
  __shared__ float tr[8][16 * 68];  // padded stride 68 -> no bank conflicts
  int wv = threadIdx.x >> 5;
  int lane = threadIdx.x & 31;
  int row0 = (blockIdx.x * 8 + wv) * 16;
  int half = lane >> 4, l15 = lane & 15;
  const float2* x1r = (const float2*)(X1 + (size_t)(row0 + l15) * 64);
  const float2* rgr = (const float2*)(right + (size_t)(row0 + l15) * 64);
  v2f afrA[16], afrB[16];
#pragma unroll
  for (int ks = 0; ks < 16; ++ks) {
    float2 ta = x1r[2 * ks + half];
    float2 tb = rgr[2 * ks + half];
    afrA[ks].x = ta.x; afrA[ks].y = ta.y;
    afrB[ks].x = tb.x; afrB[ks].y = tb.y;
  }
  v8f acc[4];
#pragma unroll
  for (int nt = 0; nt < 4; ++nt) {
    int col = nt * 16 + l15;
    float b = bo1[col];
#pragma unroll
    for (int v = 0; v < 8; ++v) acc[nt][v] = b;
#pragma unroll
    for (int ks = 0; ks < 16; ++ks) {
      int kk = 4 * ks + 2 * half;
      v2f bA, bB;
      bA.x = Wo1[kk * 64 + col];
      bA.y = Wo1[(kk + 1) * 64 + col];
      bB.x = Wo1[(64 + kk) * 64 + col];
      bB.y = Wo1[(64 + kk + 1) * 64 + col];
      acc[nt] = wmma4(afrA[ks], bA, acc[nt]);
      acc[nt] = wmma4(afrB[ks], bB, acc[nt]);
    }
  }
  float* myt = tr[wv];
#pragma unroll
  for (int nt = 0; nt < 4; ++nt)
#pragma unroll
    for (int v = 0; v < 8; ++v)
      myt[(v + half * 8) * 68 + nt * 16 + l15] = fmaxf(acc[nt][v], 0.f);
  __syncthreads();  // uniform: every wave owns one full tile
  v2f afr2[16];
#pragma unroll
  for (int ks = 0; ks < 16; ++ks) {
    int kk = 4 * ks + 2 * half;
    afr2[ks].x = myt[l15 * 68 + kk];
    afr2[ks].y = myt[l15 * 68 + kk + 1];
  }
#pragma unroll
  for (int nt = 0; nt < 4; ++nt) {
    int col = nt * 16 + l15;
    float b = bo2[col];
    v8f a2;
#pragma unroll
    for (int v = 0; v < 8; ++v) a2[v] = b;
#pragma unroll
    for (int ks = 0; ks < 16; ++ks) {
      int kk = 4 * ks + 2 * half;
      v2f bf;
      bf.x = Wo2[kk * 64 + col];
      bf.y = Wo2[(kk + 1) * 64 + col];
      a2 = wmma4(afr2[ks], bf, a2);
    }
    float* yr = right + (size_t)row0 * 64 + col;
#pragma unroll
    for (int v = 0; v < 8; ++v) yr[(size_t)(v + half * 8) * 64] = a2[v];
  }
}

// ---------------------------------------------------------------------------
// out = relu(var @ oW1 + ob1) @ oW2     (wave per node, shuffle broadcast)
// ---------------------------------------------------------------------------
__global__ __launch_bounds__(256) void k_head(
    const float* __restrict__ var, const float* __restrict__ oW1,
    const float* __restrict__ ob1, const float* __restrict__ oW2,
    float* __restrict__ out, int n) {
  int lane = threadIdx.x & 31;
  int node = blockIdx.x * 8 + (threadIdx.x >> 5);
  if (node >= n) return;
  float2 v2 = ((const float2*)(var + (size_t)node * 64))[lane];
  float ta = ob1[lane], tb = ob1[lane + 32];
#pragma unroll
  for (int d = 0; d < 64; ++d) {
    float vd = __shfl((d & 1) ? v2.y : v2.x, d >> 1);
    ta += vd * oW1[d * 64 + lane];
    tb += vd * oW1[d * 64 + lane + 32];
  }
  ta = fmaxf(ta, 0.f);
  tb = fmaxf(tb, 0.f);
  float p = ta * oW2[lane] + tb * oW2[lane + 32];
#pragma unroll
  for (int msk = 1; msk < 32; msk <<= 1) p += __shfl_xor(p, msk);
  if (lane == 0) out[node] = p;
}

// ---------------------------------------------------------------------------
extern "C" void kernel_launch(void* const* d_in, const int* in_sizes, int n_in,
                              void* d_out, int out_size, void* d_ws,
                              size_t ws_size, hipStream_t stream) {
  const float* cons_f = (const float*)d_in[0];
  // d_in[1] edge_features, d_in[20] e_lng: unused — LN over a size-1 axis
  // collapses to e_lnb, folded into the target-side bias below.
  const float* var_f = (const float*)d_in[2];
  const float* grp_f = (const float*)d_in[3];
  const float* v_lng = (const float*)d_in[4];
  const float* v_lnb = (const float*)d_in[5];
  const float* v_W1 = (const float*)d_in[6];
  const float* v_b1 = (const float*)d_in[7];
  const float* v_W2 = (const float*)d_in[8];
  const float* v_b2 = (const float*)d_in[9];
  const float* c_lng = (const float*)d_in[10];
  const float* c_lnb = (const float*)d_in[11];
  const float* c_W1 = (const float*)d_in[12];
  const float* c_b1 = (const float*)d_in[13];
  const float* c_W2 = (const float*)d_in[14];
  const float* c_b2 = (const float*)d_in[15];
  const float* g_lng = (const float*)d_in[16];
  const float* g_lnb = (const float*)d_in[17];
  const float* g_W = (const float*)d_in[18];
  const float* g_b = (const float*)d_in[19];
  const float* e_lnb = (const float*)d_in[21];
  const float* cWl = (const float*)d_in[22];
  const float* cbl = (const float*)d_in[23];
  const float* cWe = (const float*)d_in[24];
  const float* cWr = (const float*)d_in[25];
  const float* cflng = (const float*)d_in[26];
  const float* cflnb = (const float*)d_in[27];
  const float* cWf = (const float*)d_in[28];
  const float* cbf = (const float*)d_in[29];
  const float* cplng = (const float*)d_in[30];
  const float* cplnb = (const float*)d_in[31];
  const float* cWo1 = (const float*)d_in[32];
  const float* cbo1 = (const float*)d_in[33];
  const float* cWo2 = (const float*)d_in[34];
  const float* cbo2 = (const float*)d_in[35];
  const float* oW1 = (const float*)d_in[36];
  const float* ob1 = (const float*)d_in[37];
  const float* oW2 = (const float*)d_in[38];
  const int* eidx = (const int*)d_in[39];
  const int* vcol = (const int*)d_in[40];
  const int* ccol = (const int*)d_in[41];
  const int* cons_idx = eidx;
  const int* var_idx = eidx + NE;

  float* ws = (float*)d_ws;
  float* var = ws;                 // VP*64
  float* cons = var + (size_t)VP * 64;   // CP*64
  float* R = cons + (size_t)CP * 64;     // VP*64 (target-side pre + bias)
  float* L = R + (size_t)VP * 64;        // VP*64 (source-side pre)
  float* S = L + (size_t)VP * 64;        // VP*64 (segment sums)
  float* X1 = S + (size_t)VP * 64;       // VP*64 (LN(agg))
  float* degv = X1 + (size_t)VP * 64;    // VP
  float* degc = degv + VP;               // CP

  hipMemsetAsync(degv, 0, (size_t)(VP + CP) * sizeof(float), stream);
  k_deg<<<(NE + 255) / 256, 256, 0, stream>>>(var_idx, degv, NE);
  k_deg<<<(NE + 255) / 256, 256, 0, stream>>>(cons_idx, degc, NE);

  k_embed_var<<<(NV + 7) / 8, 256, 0, stream>>>(
      var_f, grp_f, vcol, v_lng, v_lnb, v_W1, v_b1, v_W2, v_b2, g_lng, g_lnb,
      g_W, g_b, var, NV);
  k_embed_cons<<<(NC + 7) / 8, 256, 0, stream>>>(
      cons_f, ccol, c_lng, c_lnb, c_W1, c_b1, c_W2, c_b2, cons, NC);

  for (int k = 0; k < 4; ++k) {
    bool to_cons = (k % 2 == 0);
    float* tgt = to_cons ? cons : var;
    float* src = to_cons ? var : cons;
    int padT = to_cons ? CP : VP;
    int padS = to_cons ? VP : CP;
    const int* ti = to_cons ? cons_idx : var_idx;
    const int* si = to_cons ? var_idx : cons_idx;
    const float* degT = to_cons ? degc : degv;

    // R = tgt @ Wl + (bl + e_lnb*We);  L = src @ Wr
    k_gemm64<<<padT / 128, 256, 0, stream>>>(tgt, cWl + k * 4096, cbl + k * 64,
                                             cWe + k * 64, e_lnb, R);
    k_gemm64<<<padS / 128, 256, 0, stream>>>(src, cWr + k * 4096, nullptr,
                                             nullptr, nullptr, L);
    hipMemsetAsync(S, 0, (size_t)padT * 64 * sizeof(float), stream);
    k_edge<<<(NE + 7) / 8, 256, 0, stream>>>(R, L, ti, si, cflng + k * 64,
                                             cflnb + k * 64, S, NE);
    k_postagg<<<padT / 128, 256, 0, stream>>>(S, degT, cWf + k * 4096,
                                              cbf + k * 64, cplng + k * 64,
                                              cplnb + k * 64, X1);
    k_outmlp<<<padT / 128, 256, 0, stream>>>(X1, tgt, cWo1 + (size_t)k * 8192,
                                             cbo1 + k * 64, cWo2 + k * 4096,
                                             cbo2 + k * 64);
  }

  k_head<<<(NV + 7) / 8, 256, 0, stream>>>(var, oW1, ob1, oW2, (float*)d_out,
                                           NV);
  (void)in_sizes; (void)n_in; (void)out_size; (void)ws_size;
}